// GraphAttentionLayer_6966436954120
// MI455X (gfx1250) — compile-verified
//
#include <hip/hip_runtime.h>
#include <math.h>

// ---------------------------------------------------------------------------
// GAT layer for MI455X (gfx1250, wave32, WMMA).
//   mapped = x @ W_map        (never materialized; reduced to sa1/sa2 in-reg)
//   value  = x @ kernel       (stored, feeds SpMM)
//   scores -> leaky -> per-row softmax -> out[i] = sum alpha * value[col] + bias
// Fused dual-GEMM uses V_WMMA_F32_16X16X4_F32 (exact fp32, matrix pipe).
// ---------------------------------------------------------------------------

typedef __attribute__((ext_vector_type(2))) float v2f;
typedef __attribute__((ext_vector_type(8))) float v8f;

#define F_DIM 256
#define D_DIM 128
#define LEAKY 0.2f

// ---------------------------------------------------------------------------
// Transpose [F,D] row-major weight -> [D,F] so WMMA B-fragments are
// K-contiguous per lane (single b64 load per fragment half).
// ---------------------------------------------------------------------------
__global__ __launch_bounds__(256) void wt_transpose(
    const float* __restrict__ W, float* __restrict__ Wt)
{
    int idx = blockIdx.x * 256 + threadIdx.x;
    if (idx >= F_DIM * D_DIM) return;
    int d = idx % D_DIM;
    int k = idx / D_DIM;
    Wt[(size_t)d * F_DIM + k] = W[idx];
}

// ---------------------------------------------------------------------------
// Fused dual GEMM. One block = 16 rows of x, 8 waves; wave w owns output
// columns [16w,16w+16) of BOTH value and (virtual) mapped.
// A-fragment (x tile) is staged in LDS once and reused by all 8 waves and
// both WMMA chains. mapped tile is immediately contracted with w1/w2.
// ---------------------------------------------------------------------------
__global__ __launch_bounds__(256) void fused_map_value(
    const float* __restrict__ x,
    const float* __restrict__ wt_map,   // [D,F]
    const float* __restrict__ wt_ker,   // [D,F]
    const float* __restrict__ w1, const float* __restrict__ b1,
    const float* __restrict__ w2, const float* __restrict__ b2,
    float* __restrict__ value,          // [N,D]
    float* __restrict__ sa1, float* __restrict__ sa2,  // [N]
    int N)
{
    __shared__ float xs[16][260];       // padded: k-loop reads hit 32 banks
    __shared__ float red1[8][16];
    __shared__ float red2[8][16];

    const int tid  = threadIdx.x;
    const int wave = tid >> 5;
    const int lane = tid & 31;
    const int half = lane >> 4;         // WMMA half-wave (K / M split)
    const int j    = lane & 15;
    const int row0 = blockIdx.x << 4;

    // Stage 16x256 fp32 x-tile (16 KB) into LDS: 4 float4 per thread.
    for (int i = 0; i < 4; ++i) {
        int idx = tid + (i << 8);       // 0..1023 float4 slots
        int r   = idx >> 6;             // 64 float4 per row
        int c4  = (idx & 63) << 2;
        int rr  = row0 + r; if (rr >= N) rr = N - 1;
        const float4 v = *(const float4*)(x + (size_t)rr * F_DIM + c4);
        *(float4*)&xs[r][c4] = v;
    }
    __syncthreads();

    const int col0 = wave << 4;
    const int col  = col0 + j;
    const float* __restrict__ wm = wt_map + (size_t)col * F_DIM;
    const float* __restrict__ wk = wt_ker + (size_t)col * F_DIM;

    v8f acc_m = {};   // mapped tile accumulator (feeds sa1/sa2 only)
    v8f acc_v = {};   // value tile accumulator

    // K-loop: 64 steps of K=4. A fragment shared by both WMMA chains.
    for (int k = 0; k < F_DIM; k += 4) {
        const int kk = k + (half << 1);             // per-half K offset
        v2f a  = *(const v2f*)&xs[j][kk];           // A[m=j][kk,kk+1]
        v2f bm = *(const v2f*)(wm + kk);            // B[kk,kk+1][n=col]
        v2f bk = *(const v2f*)(wk + kk);
        acc_m = __builtin_amdgcn_wmma_f32_16x16x4_f32(
            false, a, false, bm, (short)0, acc_m, false, false);
        acc_v = __builtin_amdgcn_wmma_f32_16x16x4_f32(
            false, a, false, bk, (short)0, acc_v, false, false);
    }

    // Store value tile: VGPR r holds rows r (lanes 0-15) / r+8 (lanes 16-31).
    // Bounds condition is uniform per block -> one scalar branch, and the
    // common path is 8 unconditional coalesced b32 stores per lane.
    if (row0 + 16 <= N) {
        for (int r = 0; r < 8; ++r) {
            int m = row0 + r + (half << 3);
            value[(size_t)m * D_DIM + col] = acc_v[r];
        }
    } else {
        for (int r = 0; r < 8; ++r) {
            int m = row0 + r + (half << 3);
            if (m < N) value[(size_t)m * D_DIM + col] = acc_v[r];
        }
    }

    // sa1/sa2: contract mapped rows with w1/w2 across the 16 columns this
    // wave owns. Shuffle-reduce within each 16-lane half (deterministic).
    const float w1c = w1[col];
    const float w2c = w2[col];
    for (int r = 0; r < 8; ++r) {
        float p1 = acc_m[r] * w1c;
        float p2 = acc_m[r] * w2c;
        for (int mms = 1; mms < 16; mms <<= 1) {
            p1 += __shfl_xor(p1, mms, 32);
            p2 += __shfl_xor(p2, mms, 32);
        }
        if (j == 0) {
            red1[wave][r + (half << 3)] = p1;
            red2[wave][r + (half << 3)] = p2;
        }
    }
    __syncthreads();

    // Fixed-order cross-wave sum (deterministic; no float atomics).
    if (tid < 16) {
        float s1 = 0.f, s2 = 0.f;
        for (int w = 0; w < 8; ++w) { s1 += red1[w][tid]; s2 += red2[w][tid]; }
        int m = row0 + tid;
        if (m < N) { sa1[m] = s1 + b1[0]; sa2[m] = s2 + b2[0]; }
    }
}

// ---------------------------------------------------------------------------
// CSR row offsets from sorted edge_row.
// ---------------------------------------------------------------------------
__global__ __launch_bounds__(256) void build_row_ptr(
    const int* __restrict__ edge_row, int* __restrict__ row_ptr, int E, int N)
{
    int e = blockIdx.x * 256 + threadIdx.x;
    if (e >= E) return;
    int r  = edge_row[e];
    int rp = (e == 0) ? -1 : edge_row[e - 1];
    for (int q = rp + 1; q <= r; ++q) row_ptr[q] = e;
    if (e == E - 1)
        for (int q = r + 1; q <= N; ++q) row_ptr[q] = E;
}

// ---------------------------------------------------------------------------
// Edge scores + leaky relu.
// ---------------------------------------------------------------------------
__global__ __launch_bounds__(256) void score_kernel(
    const float* __restrict__ adj, const float* __restrict__ sa1,
    const float* __restrict__ sa2, const int* __restrict__ edge_row,
    const int* __restrict__ edge_col, float* __restrict__ scores, int E)
{
    int e = blockIdx.x * 256 + threadIdx.x;
    if (e >= E) return;
    float a = adj[e];
    float s = a * sa1[edge_row[e]] + a * sa2[edge_col[e]];
    scores[e] = (s >= 0.f) ? s : LEAKY * s;
}

// ---------------------------------------------------------------------------
// Per-row softmax + SpMM. One wave32 per row (avg degree 16). Lane = one edge
// in the reduction phases; lane = 4 output columns (float4) in accumulate.
// value[] (51 MB) is L2-resident on MI455X (192 MB L2).
// ---------------------------------------------------------------------------
__global__ __launch_bounds__(256) void row_softmax_spmm(
    const float* __restrict__ scores, const int* __restrict__ row_ptr,
    const int* __restrict__ edge_col, const float* __restrict__ value,
    const float* __restrict__ bias, float* __restrict__ out, int N)
{
    const int lane = threadIdx.x & 31;
    const int row  = (blockIdx.x << 3) + (threadIdx.x >> 5);
    if (row >= N) return;

    const int start = row_ptr[row];
    const int end   = row_ptr[row + 1];

    float ax = 0.f, ay = 0.f, az = 0.f, aw = 0.f;
    if (start < end) {
        // pass 1: segment max
        float mx = -INFINITY;
        for (int e0 = start; e0 < end; e0 += 32) {
            int e = e0 + lane;
            float s = (e < end) ? scores[e] : -INFINITY;
            mx = fmaxf(mx, s);
        }
        for (int o = 16; o >= 1; o >>= 1) mx = fmaxf(mx, __shfl_xor(mx, o, 32));

        // pass 2: exp-sum + unnormalized accumulate, normalize at the end
        float denom = 0.f;
        for (int e0 = start; e0 < end; e0 += 32) {
            int e  = e0 + lane;
            float ex = 0.f; int c = 0;
            if (e < end) { ex = __expf(scores[e] - mx); c = edge_col[e]; }
            float t = ex;
            for (int o = 16; o >= 1; o >>= 1) t += __shfl_xor(t, o, 32);
            denom += t;
            int cnt = end - e0; if (cnt > 32) cnt = 32;
            for (int q = 0; q < cnt; ++q) {
                float aq = __shfl(ex, q, 32);
                int   cq = __shfl(c,  q, 32);
                const float4 v =
                    *(const float4*)(value + (size_t)cq * D_DIM + (lane << 2));
                ax = fmaf(aq, v.x, ax); ay = fmaf(aq, v.y, ay);
                az = fmaf(aq, v.z, az); aw = fmaf(aq, v.w, aw);
            }
        }
        float inv = 1.0f / denom;
        ax *= inv; ay *= inv; az *= inv; aw *= inv;
    }
    const float4 b = *(const float4*)(bias + (size_t)row * D_DIM + (lane << 2));
    float4 o4;
    o4.x = ax + b.x; o4.y = ay + b.y; o4.z = az + b.z; o4.w = aw + b.w;
    *(float4*)(out + (size_t)row * D_DIM + (lane << 2)) = o4;
}

// ---------------------------------------------------------------------------
// Host-side orchestration (all launches on `stream`, graph-capture safe).
// ---------------------------------------------------------------------------
extern "C" void kernel_launch(void* const* d_in, const int* in_sizes, int n_in,
                              void* d_out, int out_size, void* d_ws, size_t ws_size,
                              hipStream_t stream)
{
    const float* x     = (const float*)d_in[0];
    const float* adj   = (const float*)d_in[1];
    const float* W_map = (const float*)d_in[2];
    const float* w1    = (const float*)d_in[3];
    const float* b1    = (const float*)d_in[4];
    const float* w2    = (const float*)d_in[5];
    const float* b2    = (const float*)d_in[6];
    const float* Wker  = (const float*)d_in[7];
    const float* bias  = (const float*)d_in[8];
    const int*   erow  = (const int*)d_in[9];
    const int*   ecol  = (const int*)d_in[10];

    const int N = in_sizes[0] / F_DIM;   // 100000
    const int E = in_sizes[9];           // 1600000
    float* out = (float*)d_out;

    // Workspace layout (floats): ~59 MB total.
    float* ws      = (float*)d_ws;
    float* wt_map  = ws;                                   // D*F
    float* wt_ker  = wt_map + (size_t)F_DIM * D_DIM;       // D*F
    float* value   = wt_ker + (size_t)F_DIM * D_DIM;       // N*D
    float* sa1     = value + (size_t)N * D_DIM;            // N
    float* sa2     = sa1 + N;                              // N
    float* scores  = sa2 + N;                              // E
    int*   row_ptr = (int*)(scores + E);                   // N+1

    wt_transpose<<<(F_DIM * D_DIM + 255) / 256, 256, 0, stream>>>(W_map, wt_map);
    wt_transpose<<<(F_DIM * D_DIM + 255) / 256, 256, 0, stream>>>(Wker, wt_ker);

    fused_map_value<<<(N + 15) / 16, 256, 0, stream>>>(
        x, wt_map, wt_ker, w1, b1, w2, b2, value, sa1, sa2, N);

    build_row_ptr<<<(E + 255) / 256, 256, 0, stream>>>(erow, row_ptr, E, N);

    score_kernel<<<(E + 255) / 256, 256, 0, stream>>>(
        adj, sa1, sa2, erow, ecol, scores, E);

    row_softmax_spmm<<<(N + 7) / 8, 256, 0, stream>>>(
        scores, row_ptr, ecol, value, bias, out, N);

    (void)n_in; (void)out_size; (void)ws_size;
}